// FeaStConv_63204738728505
// MI455X (gfx1250) — compile-verified
//
#include <hip/hip_runtime.h>

typedef float v2f __attribute__((ext_vector_type(2)));
typedef float v8f __attribute__((ext_vector_type(8)));

#define NPTS   25000
#define NBATCH 2
#define KN     9      // neighbors == heads
#define FD     64
#define CD     64
#define TILE   16
#define XSTR   66     // padded LDS row stride (floats): bank = 2*r + c, conflict-free

// Low 32 bits of a generic LDS pointer are the wave-relative LDS byte address
// (flat->LDS aperture mapping: LDS address is the low 32 bits).
static __device__ __forceinline__ unsigned lds_off(const void* p) {
    return (unsigned)(uintptr_t)p;
}

__global__ __launch_bounds__(32)
void feast_conv_wmma(const float* __restrict__ x,
                     const int*   __restrict__ nbr,
                     const float* __restrict__ mlp_w,
                     const float* __restrict__ mlp_b,
                     const float* __restrict__ w_out,   // [KN*CD, FD] == w[h][c][f]
                     const float* __restrict__ bias,
                     float*       __restrict__ out)
{
    __shared__ float xn_s[KN * TILE * XSTR];   // xn_s[k][p][f], padded
    __shared__ float q_s[TILE * KN * KN];      // q_s[p][k][h]
    __shared__ int   nbr_s[TILE * KN];
    __shared__ float wsm[KN * FD];
    __shared__ float bsm[KN];

    const int lane = threadIdx.x;          // wave32
    const int r    = lane & 15;
    const int half = lane >> 4;
    const int p0   = blockIdx.x * TILE;    // 3125 tiles, exact

    // ---- stage small weights ----
    for (int t = lane; t < KN * FD; t += 32) wsm[t] = mlp_w[t];
    if (lane < KN) bsm[lane] = mlp_b[lane];

    // ---- neighbor indices (add batch offset into flattened x rows) ----
    for (int t = lane; t < TILE * KN; t += 32) {
        int p   = p0 + t / KN;
        int raw = nbr[p * KN + (t % KN)];
        int b   = p / NPTS;
        nbr_s[t] = b * NPTS + raw;
    }
    __syncthreads();

    // ---- async gather of the 16x9x64 neighbor tile straight into LDS ----
    // GLOBAL_LOAD_ASYNC_TO_LDS_B128: no VGPR round-trip, tracked by ASYNCcnt.
    for (int t = lane; t < TILE * KN * (FD / 4); t += 32) {
        int rowid = t >> 4;                 // pi*9 + k
        int fc    = (t & 15) * 4;
        int pi    = rowid / KN, k = rowid % KN;
        const float* src = x + (size_t)nbr_s[rowid] * FD + fc;
        unsigned     dst = lds_off(&xn_s[(k * TILE + pi) * XSTR + fc]);
        asm volatile("global_load_async_to_lds_b128 %0, %1, off"
                     :: "v"(dst), "v"(src) : "memory");
    }
    asm volatile("s_wait_asynccnt 0x0" ::: "memory");
    __syncthreads();

    // ---- logits + softmax over heads, per (p,k) pair ----
    for (int t = lane; t < TILE * KN; t += 32) {
        int p = t / KN, k = t % KN;
        float s[KN];
        #pragma unroll
        for (int h = 0; h < KN; ++h) s[h] = bsm[h];
        const float* xr = &xn_s[(k * TILE + p) * XSTR];
        const float* x0 = &xn_s[(0 * TILE + p) * XSTR];
        for (int f = 0; f < FD; ++f) {
            float d = xr[f] - x0[f];
            #pragma unroll
            for (int h = 0; h < KN; ++h) s[h] += d * wsm[h * FD + f];
        }
        float m = s[0];
        #pragma unroll
        for (int h = 1; h < KN; ++h) m = fmaxf(m, s[h]);
        float sum = 0.f;
        #pragma unroll
        for (int h = 0; h < KN; ++h) { s[h] = __expf(s[h] - m); sum += s[h]; }
        float inv = 1.f / sum;
        #pragma unroll
        for (int h = 0; h < KN; ++h) q_s[(p * KN + k) * KN + h] = s[h] * inv;
    }
    __syncthreads();

    // ---- accumulators pre-loaded with bias (D layout: c = nt*16 + r per lane) ----
    v8f acc0, acc1, acc2, acc3;
    {
        float b0 = bias[ 0 + r], b1 = bias[16 + r], b2 = bias[32 + r], b3 = bias[48 + r];
        #pragma unroll
        for (int j = 0; j < 8; ++j) { acc0[j] = b0; acc1[j] = b1; acc2[j] = b2; acc3[j] = b3; }
    }

    // ---- per-head: build y_h A-fragments and chain fp32 WMMAs over f (K) and c (N) ----
    for (int h = 0; h < KN; ++h) {
        float qreg[KN];
        #pragma unroll
        for (int k = 0; k < KN; ++k) qreg[k] = q_s[(r * KN + k) * KN + h];  // broadcast read

        const float* wh = w_out + (size_t)(h * CD) * FD;

        #pragma unroll
        for (int kk = 0; kk < FD / 4; ++kk) {
            const int fbase = 4 * kk + 2 * half;   // f32 A-layout: lane half selects K pair
            float a0 = 0.f, a1 = 0.f;
            #pragma unroll
            for (int k = 0; k < KN; ++k) {
                const float2 xv = *(const float2*)&xn_s[(k * TILE + r) * XSTR + fbase];
                a0 += qreg[k] * xv.x;
                a1 += qreg[k] * xv.y;
            }
            v2f A = { a0, a1 };

            {   const float* w0 = wh + (size_t)( 0 + r) * FD + fbase;
                v2f Bv = { w0[0], w0[1] };
                acc0 = __builtin_amdgcn_wmma_f32_16x16x4_f32(false, A, false, Bv, (short)0, acc0, false, false); }
            {   const float* w1 = wh + (size_t)(16 + r) * FD + fbase;
                v2f Bv = { w1[0], w1[1] };
                acc1 = __builtin_amdgcn_wmma_f32_16x16x4_f32(false, A, false, Bv, (short)0, acc1, false, false); }
            {   const float* w2 = wh + (size_t)(32 + r) * FD + fbase;
                v2f Bv = { w2[0], w2[1] };
                acc2 = __builtin_amdgcn_wmma_f32_16x16x4_f32(false, A, false, Bv, (short)0, acc2, false, false); }
            {   const float* w3 = wh + (size_t)(48 + r) * FD + fbase;
                v2f Bv = { w3[0], w3[1] };
                acc3 = __builtin_amdgcn_wmma_f32_16x16x4_f32(false, A, false, Bv, (short)0, acc3, false, false); }
        }
    }

    // ---- epilogue: relu + zero-pad mask on last point of each batch ----
    #pragma unroll
    for (int j = 0; j < 8; ++j) {
        int pm = j + 8 * half;            // D layout: VGPR j holds rows j / j+8
        int p  = p0 + pm;
        float mask = ((p % NPTS) == (NPTS - 1)) ? 0.f : 1.f;
        size_t base = (size_t)p * CD;
        out[base +  0 + r] = fmaxf(acc0[j], 0.f) * mask;
        out[base + 16 + r] = fmaxf(acc1[j], 0.f) * mask;
        out[base + 32 + r] = fmaxf(acc2[j], 0.f) * mask;
        out[base + 48 + r] = fmaxf(acc3[j], 0.f) * mask;
    }
}

extern "C" void kernel_launch(void* const* d_in, const int* in_sizes, int n_in,
                              void* d_out, int out_size, void* d_ws, size_t ws_size,
                              hipStream_t stream) {
    (void)in_sizes; (void)n_in; (void)out_size; (void)d_ws; (void)ws_size;
    const float* x     = (const float*)d_in[0];
    const int*   nbr   = (const int*)  d_in[1];
    const float* mlp_w = (const float*)d_in[2];
    const float* mlp_b = (const float*)d_in[3];
    const float* w_out = (const float*)d_in[4];
    const float* bias  = (const float*)d_in[5];
    float*       out   = (float*)d_out;

    const int total_pts = NBATCH * NPTS;        // 50000
    const int tiles     = total_pts / TILE;     // 3125 (exact)
    feast_conv_wmma<<<dim3(tiles), dim3(32), 0, stream>>>(x, nbr, mlp_w, mlp_b, w_out, bias, out);
}